// Sparsemax_66211215835414
// MI455X (gfx1250) — compile-verified
//
#include <hip/hip_runtime.h>
#include <stdint.h>

// Sparsemax over rows of a 16384 x 4096 fp32 matrix, MI455X (gfx1250).
//
// Memory-bound: 512 MiB total traffic -> ~23us at 23.3 TB/s. One block per
// row, 256 threads (8 wave32s). Row is staged global->LDS with CDNA5 async
// copies (global_load_async_to_lds_b128, ASYNCcnt / s_wait_asynccnt), then
// held in registers (16 f32/lane) for the threshold search:
//   tau solves sum(max(x - tau, 0)) == 1   (equivalent to sort+cumsum form)
// 24 bisection steps on [-1, 0] + 3 exact refinement passes.
//
// Wave reductions use ds_swizzle_b32 group-of-32 xor mode (one DS op per
// butterfly step, no VALU lane-clamp code like __shfl_xor generates).

#define ROWLEN    4096
#define NTHREADS  256
#define NWAVES    (NTHREADS / 32)
#define CHUNKS    4            // float4 chunks per thread
#define VPT       (CHUNKS * 4) // 16 floats per thread

// ds_swizzle group-of-32: offset = (xor_mask << 10) | (or_mask << 5) | and_mask
template <int XMASK>
__device__ __forceinline__ float swz_xor(float v) {
    constexpr int imm = (XMASK << 10) | 0x1f;   // and_mask=0x1f, or_mask=0
    return __int_as_float(__builtin_amdgcn_ds_swizzle(__float_as_int(v), imm));
}

__device__ __forceinline__ float wave_sum(float v) {
    v += swz_xor<16>(v);
    v += swz_xor<8>(v);
    v += swz_xor<4>(v);
    v += swz_xor<2>(v);
    v += swz_xor<1>(v);
    return v;
}

__device__ __forceinline__ float wave_max(float v) {
    v = fmaxf(v, swz_xor<16>(v));
    v = fmaxf(v, swz_xor<8>(v));
    v = fmaxf(v, swz_xor<4>(v));
    v = fmaxf(v, swz_xor<2>(v));
    v = fmaxf(v, swz_xor<1>(v));
    return v;
}

__global__ __launch_bounds__(NTHREADS) void
sparsemax_rows_kernel(const float* __restrict__ x, float* __restrict__ out, int nrows) {
    __shared__ float srow[ROWLEN];                    // 16 KB staging for one row
    __shared__ __align__(16) float sredA[NWAVES];
    __shared__ __align__(16) float sredB[NWAVES];

    const int tid  = threadIdx.x;
    const int wave = tid >> 5;
    const int lane = tid & 31;
    const int row  = blockIdx.x;
    if (row >= nrows) return;                         // uniform guard

    const float* rowp = x + (size_t)row * ROWLEN;

    // ---- async copy: global row -> LDS (ASYNCcnt-tracked, no VGPR data) ----
    // Low 32 bits of the flat address of a __shared__ object are the
    // workgroup-relative LDS byte address (ISA 10.2 aperture mapping).
    const uint32_t lds_base = (uint32_t)(uintptr_t)&srow[0];
    #pragma unroll
    for (int j = 0; j < CHUNKS; ++j) {
        const int chunk = tid + j * NTHREADS;         // 0..1023, coalesced
        uint32_t lds_addr = lds_base + (uint32_t)chunk * 16u;
        uint64_t gaddr    = (uint64_t)(uintptr_t)(rowp + chunk * 4);
        asm volatile("global_load_async_to_lds_b128 %0, %1, off"
                     :: "v"(lds_addr), "v"(gaddr) : "memory");
    }
    asm volatile("s_wait_asynccnt 0" ::: "memory");
    __syncthreads();

    // ---- pull 16 elements/lane from LDS into registers (ds_load_b128) ----
    float v[VPT];
    #pragma unroll
    for (int j = 0; j < CHUNKS; ++j) {
        const int chunk = tid + j * NTHREADS;
        float4 q = *reinterpret_cast<const float4*>(&srow[chunk * 4]);
        v[j * 4 + 0] = q.x; v[j * 4 + 1] = q.y;
        v[j * 4 + 2] = q.z; v[j * 4 + 3] = q.w;
    }

    // ---- row max, then shift so max == 0 ----
    float m = v[0];
    #pragma unroll
    for (int j = 1; j < VPT; ++j) m = fmaxf(m, v[j]);
    m = wave_max(m);
    if (lane == 0) sredA[wave] = m;
    __syncthreads();
    {
        float4 a0 = *reinterpret_cast<const float4*>(&sredA[0]);
        float4 a1 = *reinterpret_cast<const float4*>(&sredA[4]);
        float rmax = fmaxf(fmaxf(fmaxf(a0.x, a0.y), fmaxf(a0.z, a0.w)),
                           fmaxf(fmaxf(a1.x, a1.y), fmaxf(a1.z, a1.w)));
        #pragma unroll
        for (int j = 0; j < VPT; ++j) v[j] -= rmax;
    }
    __syncthreads();

    // ---- bisection: f(tau) = sum(max(v - tau, 0)) is monotone decreasing,
    //      f(-1) >= 1 >= f(0) after the shift, so root is in [-1, 0]. ----
    float lo = -1.0f, hi = 0.0f;
    for (int it = 0; it < 24; ++it) {
        const float mid = 0.5f * (lo + hi);
        float p = 0.0f;
        #pragma unroll
        for (int j = 0; j < VPT; ++j) p += fmaxf(v[j] - mid, 0.0f);
        p = wave_sum(p);
        if (lane == 0) sredA[wave] = p;
        __syncthreads();
        float4 a0 = *reinterpret_cast<const float4*>(&sredA[0]);
        float4 a1 = *reinterpret_cast<const float4*>(&sredA[4]);
        float s = ((a0.x + a0.y) + (a0.z + a0.w)) + ((a1.x + a1.y) + (a1.z + a1.w));
        __syncthreads();
        if (s >= 1.0f) lo = mid; else hi = mid;       // uniform across block
    }

    // ---- exact refinement on the identified linear segment:
    //      tau = (sum_{v > tau} v - 1) / |{v > tau}| ----
    float tau = 0.5f * (lo + hi);
    for (int pass = 0; pass < 3; ++pass) {
        float ps = 0.0f, pc = 0.0f;
        #pragma unroll
        for (int j = 0; j < VPT; ++j) {
            if (v[j] > tau) { ps += v[j]; pc += 1.0f; }
        }
        ps = wave_sum(ps);
        pc = wave_sum(pc);
        if (lane == 0) { sredA[wave] = ps; sredB[wave] = pc; }
        __syncthreads();
        float4 a0 = *reinterpret_cast<const float4*>(&sredA[0]);
        float4 a1 = *reinterpret_cast<const float4*>(&sredA[4]);
        float4 b0 = *reinterpret_cast<const float4*>(&sredB[0]);
        float4 b1 = *reinterpret_cast<const float4*>(&sredB[4]);
        float s = ((a0.x + a0.y) + (a0.z + a0.w)) + ((a1.x + a1.y) + (a1.z + a1.w));
        float c = ((b0.x + b0.y) + (b0.z + b0.w)) + ((b1.x + b1.y) + (b1.z + b1.w));
        __syncthreads();
        tau = (s - 1.0f) / c;     // c >= 1 (max elem is 0 > tau), tau < 0 always
    }

    // ---- write out = max(v - tau, 0), coalesced float4 stores ----
    float* orow = out + (size_t)row * ROWLEN;
    #pragma unroll
    for (int j = 0; j < CHUNKS; ++j) {
        const int chunk = tid + j * NTHREADS;
        float4 q;
        q.x = fmaxf(v[j * 4 + 0] - tau, 0.0f);
        q.y = fmaxf(v[j * 4 + 1] - tau, 0.0f);
        q.z = fmaxf(v[j * 4 + 2] - tau, 0.0f);
        q.w = fmaxf(v[j * 4 + 3] - tau, 0.0f);
        *reinterpret_cast<float4*>(orow + chunk * 4) = q;
    }
}

extern "C" void kernel_launch(void* const* d_in, const int* in_sizes, int n_in,
                              void* d_out, int out_size, void* d_ws, size_t ws_size,
                              hipStream_t stream) {
    const float* x = (const float*)d_in[0];
    float* out = (float*)d_out;
    const int nrows = in_sizes[0] / ROWLEN;   // 16384
    dim3 grid(nrows), block(NTHREADS);
    sparsemax_rows_kernel<<<grid, block, 0, stream>>>(x, out, nrows);
}